// MixtralDecodeAttention_10101763080526
// MI455X (gfx1250) — compile-verified
//
#include <hip/hip_runtime.h>
#include <math.h>

// ---------------------------------------------------------------------------
// Mixtral decode attention, MI455X (gfx1250), wave32 + WMMA f32 16x16x4.
// Memory-bound (~1.25 GB @ 23.3 TB/s => ~54us floor). fp32 end-to-end.
// ---------------------------------------------------------------------------

#define HIDDEN    4096
#define N_HEADS   32
#define N_KV      8
#define N_REP     4
#define HEAD_DIM  128
#define CACHE_LEN 4096
#define BATCH     32
#define N_SPLITS  8
#define SPLIT_LEN (CACHE_LEN / N_SPLITS)   // 512

typedef float v2f __attribute__((ext_vector_type(2)));
typedef float v8f __attribute__((ext_vector_type(8)));

__device__ __forceinline__ v8f wmma_f32(v2f a, v2f b, v8f c) {
  // 8-arg pattern: (neg_a, A, neg_b, B, c_mod, C, reuse_a, reuse_b)
  return __builtin_amdgcn_wmma_f32_16x16x4_f32(false, a, false, b, (short)0, c,
                                               false, false);
}

__device__ __forceinline__ float wave_max32(float v) {
  #pragma unroll
  for (int o = 16; o > 0; o >>= 1) v = fmaxf(v, __shfl_xor(v, o, 32));
  return v;
}
__device__ __forceinline__ float wave_sum32(float v) {
  #pragma unroll
  for (int o = 16; o > 0; o >>= 1) v += __shfl_xor(v, o, 32);
  return v;
}

// ---------------------------------------------------------------------------
// GEMM: C[32 x N] += A[32 x K] * B[K x N]  (A,B,C row-major, fp32)
// One wave per 16-wide N tile; both 16-row M tiles per wave so B is read once.
// grid.x = N/16 tiles, grid.y = K-splits. Accumulate with fp32 atomics.
// N/K compile-time so every load in the unrolled body is base+imm_offset.
// ---------------------------------------------------------------------------
template <int N, int K, int KCHUNK>
__global__ __launch_bounds__(32) void gemm32_wmma(
    const float* __restrict__ A, const float* __restrict__ B,
    float* __restrict__ C) {
  const int lane = threadIdx.x;
  const int n0   = blockIdx.x * 16;
  const int kb   = blockIdx.y * KCHUNK;
  const int koff = (lane >= 16) ? 2 : 0;
  const int n16  = lane & 15;

  v8f c0 = {}; // rows 0..15
  v8f c1 = {}; // rows 16..31

  const float* a0p = A + (size_t)n16 * K + kb + koff;        // row n16
  const float* a1p = a0p + (size_t)16 * K;                   // row n16+16
  const float* bp  = B + (size_t)(kb + koff) * N + n0 + n16; // row kb+koff

  #pragma unroll
  for (int blk = 0; blk < KCHUNK / 32; ++blk) {
    #pragma unroll
    for (int u = 0; u < 8; ++u) {
      v2f aF0 = *(const v2f*)(a0p + u * 4);
      v2f aF1 = *(const v2f*)(a1p + u * 4);
      v2f bF;
      bF.x = bp[(size_t)(u * 4) * N];       // row kb+koff+4u   (imm offset)
      bF.y = bp[(size_t)(u * 4 + 1) * N];   // row kb+koff+4u+1 (imm offset)
      c0 = wmma_f32(aF0, bF, c0);
      c1 = wmma_f32(aF1, bF, c1);
    }
    a0p += 32;
    a1p += 32;
    bp  += (size_t)32 * N;
  }

  // D layout: VGPR j = row j (lanes 0-15) / row j+8 (lanes 16-31), col = lane&15
  float* cp = C + n0 + n16 + ((lane >= 16) ? (size_t)8 * N : (size_t)0);
  #pragma unroll
  for (int j = 0; j < 8; ++j) {
    unsafeAtomicAdd(cp + (size_t)j * N, c0[j]);
    unsafeAtomicAdd(cp + (size_t)(j + 16) * N, c1[j]);
  }
}

// ---------------------------------------------------------------------------
// Flash-decoding attention split. One wave per (b, kv, split).
// Scores via WMMA f32 16x16x4 (rows 0-3 = the 4 GQA reps; rows 4-15 are
// replicated padding, never reduced). q A-fragments are k-tile-invariant and
// held in registers. Online softmax, coalesced float4 P*V.
// ---------------------------------------------------------------------------
__global__ __launch_bounds__(32) void attn_split_kernel(
    const float* __restrict__ q, const float* __restrict__ kc,
    const float* __restrict__ vc, float* __restrict__ pacc,
    float* __restrict__ pml) {
  const int lane = threadIdx.x;
  const int wg   = blockIdx.x;              // b*64 + kv*8 + s
  const int s    = wg & 7;
  const int kv   = (wg >> 3) & 7;
  const int b    = wg >> 6;

  const int   koff  = (lane >= 16) ? 2 : 0;
  const int   r4    = lane & 3;             // padding rows replicate rows 0-3
  const int   n16   = lane & 15;
  const float scale = 0.08838834764831845f; // 1/sqrt(128)

  // Preload this lane's q A-fragments for all 32 k-steps (row r4, 2 floats per
  // step). Same address across replicated lanes -> broadcast from cache.
  const float* qg = q + (size_t)b * HIDDEN + (size_t)kv * N_REP * HEAD_DIM +
                    (size_t)r4 * HEAD_DIM + koff;
  v2f qreg[HEAD_DIM / 4];
  #pragma unroll
  for (int i = 0; i < HEAD_DIM / 4; ++i) qreg[i] = *(const v2f*)(qg + i * 4);

  const size_t bk    = (size_t)(b * N_KV + kv);
  const float* kbase = kc + (bk * CACHE_LEN + (size_t)s * SPLIT_LEN) * HEAD_DIM;
  const float* vbase = vc + (bk * CACHE_LEN + (size_t)s * SPLIT_LEN) * HEAD_DIM;

  float  m[4] = {-INFINITY, -INFINITY, -INFINITY, -INFINITY};
  float  l[4] = {0.f, 0.f, 0.f, 0.f};
  float4 acc[4] = {};

  const float* krow = kbase + (size_t)n16 * HEAD_DIM + koff;  // lane's key row
  const float* vrow = vbase + (size_t)lane * 4;               // lane's dims

  for (int t = 0; t < SPLIT_LEN; t += 16) {
    // ---- scores D(16x16) = Q(16x128) * K_tile^T via 32 x wmma 16x16x4 ----
    v8f d = {};
    #pragma unroll
    for (int i = 0; i < HEAD_DIM / 4; ++i) {
      v2f bF = *(const v2f*)(krow + i * 4);
      d = wmma_f32(qreg[i], bF, d);
    }
    krow += (size_t)16 * HEAD_DIM;

    // ---- online softmax over the 4 real rows ----
    float p[4];
    #pragma unroll
    for (int r = 0; r < 4; ++r) {
      float sc   = (lane < 16) ? d[r] * scale : -INFINITY;
      float tmax = wave_max32(sc);
      float mn   = fmaxf(m[r], tmax);
      float corr = __expf(m[r] - mn);
      p[r]       = (lane < 16) ? __expf(sc - mn) : 0.0f;
      float ts   = wave_sum32(p[r]);
      l[r] = l[r] * corr + ts;
      m[r] = mn;
      acc[r].x *= corr; acc[r].y *= corr; acc[r].z *= corr; acc[r].w *= corr;
    }

    // ---- P*V: lane owns dims [4*lane .. 4*lane+3]; V row loads coalesce ----
    #pragma unroll
    for (int j = 0; j < 16; ++j) {
      float4 v4 = *(const float4*)(vrow + (size_t)j * HEAD_DIM);
      float p0 = __shfl(p[0], j, 32);
      float p1 = __shfl(p[1], j, 32);
      float p2 = __shfl(p[2], j, 32);
      float p3 = __shfl(p[3], j, 32);
      acc[0].x += p0 * v4.x; acc[0].y += p0 * v4.y; acc[0].z += p0 * v4.z; acc[0].w += p0 * v4.w;
      acc[1].x += p1 * v4.x; acc[1].y += p1 * v4.y; acc[1].z += p1 * v4.z; acc[1].w += p1 * v4.w;
      acc[2].x += p2 * v4.x; acc[2].y += p2 * v4.y; acc[2].z += p2 * v4.z; acc[2].w += p2 * v4.w;
      acc[3].x += p3 * v4.x; acc[3].y += p3 * v4.y; acc[3].z += p3 * v4.z; acc[3].w += p3 * v4.w;
    }
    vrow += (size_t)16 * HEAD_DIM;
  }

  // ---- write split partials ----
  float* ab = pacc + ((bk * N_SPLITS + s) * N_REP) * HEAD_DIM;
  #pragma unroll
  for (int r = 0; r < 4; ++r)
    *(float4*)(ab + (size_t)r * HEAD_DIM + lane * 4) = acc[r];
  if (lane == 0) {
    float* mlp = pml + (bk * N_SPLITS + s) * 8;
    #pragma unroll
    for (int r = 0; r < 4; ++r) { mlp[r] = m[r]; mlp[4 + r] = l[r]; }
  }
}

// ---------------------------------------------------------------------------
// Combine splits (log-sum-exp merge) + add v_new broadcast.
// One block per (b, kv); thread = head dim.
// ---------------------------------------------------------------------------
__global__ __launch_bounds__(128) void attn_reduce_kernel(
    const float* __restrict__ pacc, const float* __restrict__ pml,
    const float* __restrict__ vnew, float* __restrict__ attn) {
  const int d  = threadIdx.x;
  const int kv = blockIdx.x & 7;
  const int b  = blockIdx.x >> 3;
  const size_t bk = (size_t)(b * N_KV + kv);

  __shared__ float ml[N_SPLITS * 8];
  if (d < N_SPLITS * 8) ml[d] = pml[bk * N_SPLITS * 8 + d];
  __syncthreads();

  const float vn = vnew[(size_t)b * (N_KV * HEAD_DIM) + kv * HEAD_DIM + d];
  #pragma unroll
  for (int r = 0; r < 4; ++r) {
    float M = -INFINITY;
    #pragma unroll
    for (int s = 0; s < N_SPLITS; ++s) M = fmaxf(M, ml[s * 8 + r]);
    float L = 0.f, o = 0.f;
    #pragma unroll
    for (int s = 0; s < N_SPLITS; ++s) {
      float w = __expf(ml[s * 8 + r] - M);
      L += ml[s * 8 + 4 + r] * w;
      o += pacc[((bk * N_SPLITS + s) * N_REP + r) * HEAD_DIM + d] * w;
    }
    attn[(size_t)b * HIDDEN + (kv * N_REP + r) * HEAD_DIM + d] = o / L + vn;
  }
}

// ---------------------------------------------------------------------------
extern "C" void kernel_launch(void* const* d_in, const int* in_sizes, int n_in,
                              void* d_out, int out_size, void* d_ws,
                              size_t ws_size, hipStream_t stream) {
  const float* x  = (const float*)d_in[0];  // 32 x 1 x 4096
  const float* kc = (const float*)d_in[1];  // 32 x 8 x 4096 x 128
  const float* vc = (const float*)d_in[2];
  const float* wq = (const float*)d_in[3];  // 4096 x 4096
  // d_in[4] = wk : computed by the reference but never used in the output.
  const float* wv = (const float*)d_in[5];  // 4096 x 1024
  const float* wo = (const float*)d_in[6];  // 4096 x 4096
  float* out = (float*)d_out;               // 32 x 4096

  // workspace layout (floats)
  float* qbuf = (float*)d_ws;                          // 32*4096
  float* vnew = qbuf + (size_t)BATCH * HIDDEN;         // 32*1024
  float* attn = vnew + (size_t)BATCH * N_KV * HEAD_DIM;// 32*4096
  float* pacc = attn + (size_t)BATCH * HIDDEN;         // 32*8*8*4*128
  float* pml  = pacc + (size_t)BATCH * N_KV * N_SPLITS * N_REP * HEAD_DIM;

  // zero the atomic-accumulated buffers
  hipMemsetAsync(qbuf, 0,
                 ((size_t)BATCH * HIDDEN + (size_t)BATCH * N_KV * HEAD_DIM) *
                     sizeof(float), stream);
  hipMemsetAsync(out, 0, (size_t)BATCH * HIDDEN * sizeof(float), stream);

  // Q = x @ wq  (N=4096), Vnew = x @ wv (N=1024); K split 4x for occupancy
  gemm32_wmma<HIDDEN, HIDDEN, HIDDEN / 4>
      <<<dim3(HIDDEN / 16, 4), 32, 0, stream>>>(x, wq, qbuf);
  gemm32_wmma<N_KV * HEAD_DIM, HIDDEN, HIDDEN / 4>
      <<<dim3((N_KV * HEAD_DIM) / 16, 4), 32, 0, stream>>>(x, wv, vnew);

  // flash-decoding over the KV cache: 32b * 8kv * 8 splits = 2048 waves
  attn_split_kernel<<<BATCH * N_KV * N_SPLITS, 32, 0, stream>>>(qbuf, kc, vc,
                                                                pacc, pml);
  attn_reduce_kernel<<<BATCH * N_KV, 128, 0, stream>>>(pacc, pml, vnew, attn);

  // out = attn @ wo
  gemm32_wmma<HIDDEN, HIDDEN, HIDDEN / 4>
      <<<dim3(HIDDEN / 16, 4), 32, 0, stream>>>(attn, wo, out);
}